// Se3FlowMatching_65635690217862
// MI455X (gfx1250) — compile-verified
//
#include <hip/hip_runtime.h>
#include <hip/hip_bf16.h>
#include <math.h>

typedef __attribute__((ext_vector_type(2))) float v2f;
typedef __attribute__((ext_vector_type(8))) float v8f;

#define NN 1024
#define HH 256
#define EHD 128
#define KP 264   // padded K for [h | delta_local | t_emb] = 262 -> 264

__device__ __forceinline__ float silu_f(float x) { return x / (1.0f + __expf(-x)); }

// ---------------------------------------------------------------------------
// small 3-vector helpers
struct F3 { float x, y, z; };
__device__ __forceinline__ F3 f3add(F3 a, F3 b) { return {a.x + b.x, a.y + b.y, a.z + b.z}; }
__device__ __forceinline__ F3 f3sub(F3 a, F3 b) { return {a.x - b.x, a.y - b.y, a.z - b.z}; }
__device__ __forceinline__ F3 f3scale(F3 a, float s) { return {a.x * s, a.y * s, a.z * s}; }
__device__ __forceinline__ float f3dot(F3 a, F3 b) { return a.x * b.x + a.y * b.y + a.z * b.z; }
__device__ __forceinline__ F3 f3cross(F3 a, F3 b) {
  return {a.y * b.z - a.z * b.y, a.z * b.x - a.x * b.z, a.x * b.y - a.y * b.x};
}
__device__ __forceinline__ F3 safedir(F3 p, F3 f) {
  // _safe_direction(primary, fallback, eps=1e-6) from the reference
  float ss = f3dot(p, p);
  float nrm = sqrtf(fmaxf(ss, 1e-6f));
  F3 o = (nrm < 1e-6f) ? f : p;          // (always p given the clip, kept for fidelity)
  float os = f3dot(o, o);
  return f3scale(o, 1.0f / sqrtf(fmaxf(os, 1e-6f)));
}

// ---------------------------------------------------------------------------
// 1) per-node frames + delta_local
__global__ void frames_kernel(const float* __restrict__ xc, const float* __restrict__ xt,
                              float* __restrict__ frames, float* __restrict__ dl) {
  int n = blockIdx.x * blockDim.x + threadIdx.x;
  if (n >= NN) return;
  auto ld = [&](int i) { return F3{xc[i * 3 + 0], xc[i * 3 + 1], xc[i * 3 + 2]}; };
  F3 prev = (n >= 1)    ? f3sub(ld(n), ld(n - 1))       : f3sub(ld(1), ld(0));
  F3 nxt  = (n < NN - 1)? f3sub(ld(n + 1), ld(n))       : f3sub(ld(NN - 1), ld(NN - 2));
  F3 tangent = safedir(f3add(prev, nxt), nxt);
  F3 curv    = safedir(f3sub(nxt, prev), prev);
  F3 ayr     = f3sub(curv, f3scale(tangent, f3dot(curv, tangent)));
  F3 ay      = safedir(ayr, f3cross(tangent, prev));
  F3 az      = safedir(f3cross(tangent, ay), f3cross(tangent, curv));
  ay         = safedir(f3cross(az, tangent), ay);
  float* Fr = frames + n * 9;                 // row-major [3][3], columns = (tangent, ay, az)
  Fr[0] = tangent.x; Fr[1] = ay.x; Fr[2] = az.x;
  Fr[3] = tangent.y; Fr[4] = ay.y; Fr[5] = az.y;
  Fr[6] = tangent.z; Fr[7] = ay.z; Fr[8] = az.z;
  F3 d = {xt[n * 3 + 0] - xc[n * 3 + 0], xt[n * 3 + 1] - xc[n * 3 + 1],
          xt[n * 3 + 2] - xc[n * 3 + 2]};
  dl[n * 3 + 0] = f3dot(tangent, d);
  dl[n * 3 + 1] = f3dot(ay, d);
  dl[n * 3 + 2] = f3dot(az, d);
}

// ---------------------------------------------------------------------------
// 2) pad + transpose weight [Krows,Nc] -> Wt[Nc,Kp] so B fragments are one
//    contiguous 8-byte load in the GEMM hot loop. Rows >= Krows are zero.
__global__ void transpose_pad_kernel(const float* __restrict__ W, float* __restrict__ Wt,
                                     int Krows, int Kp, int Nc) {
  int idx = blockIdx.x * blockDim.x + threadIdx.x;
  if (idx >= Nc * Kp) return;
  int col = idx / Kp, k = idx % Kp;
  Wt[idx] = (k < Krows) ? W[k * Nc + col] : 0.0f;
}

// ---------------------------------------------------------------------------
// 3) assemble node_feat[N,264] = [h | delta_local | (t, sin t, cos t) | 0,0]
__global__ void nodefeat_kernel(const float* __restrict__ h, const float* __restrict__ dl,
                                const float* __restrict__ tau, float* __restrict__ nf) {
  int idx = blockIdx.x * blockDim.x + threadIdx.x;
  if (idx >= NN * KP) return;
  int n = idx / KP, k = idx % KP;
  float v;
  if (k < HH)           v = h[n * HH + k];
  else if (k < HH + 3)  v = dl[n * 3 + (k - HH)];
  else if (k == HH + 3) v = tau[0];
  else if (k == HH + 4) v = __sinf(tau[0]);
  else if (k == HH + 5) v = __cosf(tau[0]);
  else                  v = 0.0f;
  nf[idx] = v;
}

// assemble uf[N,264] = [nh | delta_local | msgs | 0,0]
__global__ void uf_kernel(const float* __restrict__ nh, const float* __restrict__ dl,
                          const float* __restrict__ msgs, float* __restrict__ uf) {
  int idx = blockIdx.x * blockDim.x + threadIdx.x;
  if (idx >= NN * KP) return;
  int n = idx / KP, k = idx % KP;
  float v;
  if (k < HH)           v = nh[n * HH + k];
  else if (k < HH + 3)  v = dl[n * 3 + (k - HH)];
  else if (k < HH + 6)  v = msgs[n * 3 + (k - HH - 3)];
  else                  v = 0.0f;
  uf[idx] = v;
}

// ---------------------------------------------------------------------------
// 4) fused GEMM + bias + SiLU: C[M,Nc] = silu(A[M,K] @ W + b), W pre-transposed
//    to Wt[Nc,K]. WMMA f32 16x16x4, bias folded into the accumulator init.
//    One wave owns a 16x64 output strip (A fragment reused 4x per K-step).
//    A fragment (16x4 f32): lane l -> M = l%16, VGPR v -> K = v + 2*(l/16)
//    B fragment (4x16 f32): lane l -> N = l%16, VGPR v -> K = v + 2*(l/16)
//    D (16x16 f32): lane l -> N = l%16, VGPR v -> M = v + 8*(l/16)
__global__ void wmma_gemm_silu(const float* __restrict__ A, const float* __restrict__ Wt,
                               const float* __restrict__ bias, float* __restrict__ C,
                               int M, int Nc, int K) {
  int lane = threadIdx.x & 31;
  int wave = blockIdx.x * (blockDim.x >> 5) + (threadIdx.x >> 5);
  int strips = Nc >> 6;
  int tm = wave / strips;
  int tn = (wave % strips) << 6;
  if (tm * 16 >= M) return;                       // wave-uniform
  int cl = lane & 15, khalf = (lane >> 4) << 1;   // 0 or 2
  int row = tm * 16 + cl;
  v8f acc[4];
#pragma unroll
  for (int j = 0; j < 4; ++j) {
    float bb = bias[tn + j * 16 + cl];            // same column for all 8 row-VGPRs
#pragma unroll
    for (int v = 0; v < 8; ++v) acc[j][v] = bb;
  }
  for (int k0 = 0; k0 < K; k0 += 4) {
    v2f a = *reinterpret_cast<const v2f*>(A + row * K + k0 + khalf);
#pragma unroll
    for (int j = 0; j < 4; ++j) {
      int col = tn + j * 16 + cl;
      v2f b = *reinterpret_cast<const v2f*>(Wt + col * K + k0 + khalf);
      acc[j] = __builtin_amdgcn_wmma_f32_16x16x4_f32(false, a, false, b,
                                                     (short)0, acc[j], false, false);
    }
  }
  int rbase = tm * 16 + ((lane >> 4) << 3);
#pragma unroll
  for (int j = 0; j < 4; ++j) {
    int col = tn + j * 16 + cl;
#pragma unroll
    for (int v = 0; v < 8; ++v)
      C[(rbase + v) * Nc + col] = silu_f(acc[j][v]);
  }
}

// ---------------------------------------------------------------------------
// 5) src/dst/ns heads: wave per row, shuffle reduction
__global__ void heads_kernel(const float* __restrict__ nh,
                             const float* __restrict__ w_src, const float* __restrict__ b_src,
                             const float* __restrict__ w_dst, const float* __restrict__ b_dst,
                             const float* __restrict__ w_ns, const float* __restrict__ b_ns,
                             float* __restrict__ src, float* __restrict__ dst,
                             float* __restrict__ ns) {
  int lane = threadIdx.x & 31;
  int n = blockIdx.x * (blockDim.x >> 5) + (threadIdx.x >> 5);
  if (n >= NN) return;
  float s1 = 0.f, s2 = 0.f, s3 = 0.f;
  for (int c = lane; c < HH; c += 32) {
    float v = nh[n * HH + c];
    s1 += v * w_src[c]; s2 += v * w_dst[c]; s3 += v * w_ns[c];
  }
#pragma unroll
  for (int off = 16; off > 0; off >>= 1) {
    s1 += __shfl_xor(s1, off, 32);
    s2 += __shfl_xor(s2, off, 32);
    s3 += __shfl_xor(s3, off, 32);
  }
  if (lane == 0) {
    src[n] = s1 + b_src[0];
    dst[n] = s2 + b_dst[0];
    ns[n]  = 1.0f / (1.0f + __expf(-(s3 + b_ns[0])));
  }
}

// ---------------------------------------------------------------------------
// 6) edge MLP + logits. One wave per node n; loops over 64 tiles of 16 m's.
//    Per tile: 8x V_WMMA_F32_16X16X4_F32 (EH=128 in 16-chunks) with b_eg1
//    folded into the accumulator init, fused SiLU + dot(w_eg2), cross-lane
//    reduce, masked store of the logits row.
__global__ void edge_logits_kernel(const float* __restrict__ xt, const float* __restrict__ frames,
                                   const float* __restrict__ W_eg1, const float* __restrict__ b_eg1,
                                   const float* __restrict__ w_eg2, const float* __restrict__ b_eg2,
                                   const float* __restrict__ src, const float* __restrict__ dst,
                                   float* __restrict__ logits) {
  int lane = threadIdx.x & 31;
  int n = blockIdx.x * (blockDim.x >> 5) + (threadIdx.x >> 5);
  if (n >= NN) return;                              // wave-uniform
  int cl = lane & 15, hi = lane >> 4, khalf = hi << 1;
  float F0 = frames[n * 9 + 0], F1 = frames[n * 9 + 1], F2 = frames[n * 9 + 2];
  float F3_ = frames[n * 9 + 3], F4 = frames[n * 9 + 4], F5 = frames[n * 9 + 5];
  float F6 = frames[n * 9 + 6], F7 = frames[n * 9 + 7], F8 = frames[n * 9 + 8];
  float xnx = xt[n * 3 + 0], xny = xt[n * 3 + 1], xnz = xt[n * 3 + 2];
  float srcn = src[n], b2 = b_eg2[0];
  v2f bf[8];
  float be[8], w2[8];
#pragma unroll
  for (int t = 0; t < 8; ++t) {
    int ch = t * 16 + cl;
    bf[t].x = W_eg1[khalf * EHD + ch];
    bf[t].y = W_eg1[(khalf + 1) * EHD + ch];
    be[t] = b_eg1[ch];
    w2[t] = w_eg2[ch];
  }
  for (int mt = 0; mt < NN / 16; ++mt) {
    int m = mt * 16 + cl;
    float rx = xt[m * 3 + 0] - xnx, ry = xt[m * 3 + 1] - xny, rz = xt[m * 3 + 2] - xnz;
    float rlx = F0 * rx + F3_ * ry + F6 * rz;       // frames^T @ rel
    float rly = F1 * rx + F4 * ry + F7 * rz;
    float rlz = F2 * rx + F5 * ry + F8 * rz;
    float dist = sqrtf(rlx * rlx + rly * rly + rlz * rlz);
    v2f a;
    a.x = hi ? rlz : rlx;                           // A 16x4: lanes0-15 K=0,1; 16-31 K=2,3
    a.y = hi ? dist : rly;
    float part[8] = {0.f, 0.f, 0.f, 0.f, 0.f, 0.f, 0.f, 0.f};
#pragma unroll
    for (int t = 0; t < 8; ++t) {
      v8f d;                                        // bias pre-loaded into accumulator
#pragma unroll
      for (int v = 0; v < 8; ++v) d[v] = be[t];
      d = __builtin_amdgcn_wmma_f32_16x16x4_f32(false, a, false, bf[t],
                                                (short)0, d, false, false);
#pragma unroll
      for (int v = 0; v < 8; ++v)
        part[v] += silu_f(d[v]) * w2[t];
    }
#pragma unroll
    for (int v = 0; v < 8; ++v) {                   // reduce the 16 channels-per-half
      float s = part[v];
      s += __shfl_xor(s, 1, 16);
      s += __shfl_xor(s, 2, 16);
      s += __shfl_xor(s, 4, 16);
      s += __shfl_xor(s, 8, 16);
      part[v] = s;
    }
    if (cl == 0) {
#pragma unroll
      for (int v = 0; v < 8; ++v) {
        int mrow = mt * 16 + v + 8 * hi;
        float lg = part[v] + srcn + dst[mrow] + b2;
        if (mrow == n) lg = -1e30f;                 // diag masked (exp underflows to 0)
        logits[n * NN + mrow] = lg;
      }
    }
  }
}

// ---------------------------------------------------------------------------
// 7) row softmax + attention-weighted message aggregation
__global__ void softmax_msgs_kernel(const float* __restrict__ logits, const float* __restrict__ xt,
                                    const float* __restrict__ frames, const float* __restrict__ ns,
                                    float* __restrict__ msgs) {
  __shared__ float red[256];
  int n = blockIdx.x, t = threadIdx.x;
  float F0 = frames[n * 9 + 0], F1 = frames[n * 9 + 1], F2 = frames[n * 9 + 2];
  float F3_ = frames[n * 9 + 3], F4 = frames[n * 9 + 4], F5 = frames[n * 9 + 5];
  float F6 = frames[n * 9 + 6], F7 = frames[n * 9 + 7], F8 = frames[n * 9 + 8];
  float xnx = xt[n * 3 + 0], xny = xt[n * 3 + 1], xnz = xt[n * 3 + 2];
  const float* row = logits + n * NN;
  float lmax = -3.0e38f;
  for (int m = t; m < NN; m += 256) lmax = fmaxf(lmax, row[m]);
  red[t] = lmax; __syncthreads();
  for (int s = 128; s > 0; s >>= 1) { if (t < s) red[t] = fmaxf(red[t], red[t + s]); __syncthreads(); }
  float rmax = red[0]; __syncthreads();
  float sum = 0.f, sx = 0.f, sy = 0.f, sz = 0.f;
  for (int m = t; m < NN; m += 256) {
    float p = __expf(row[m] - rmax);
    float rx = xt[m * 3 + 0] - xnx, ry = xt[m * 3 + 1] - xny, rz = xt[m * 3 + 2] - xnz;
    float rlx = F0 * rx + F3_ * ry + F6 * rz;
    float rly = F1 * rx + F4 * ry + F7 * rz;
    float rlz = F2 * rx + F5 * ry + F8 * rz;
    float w = p * ns[m];
    sum += p; sx += w * rlx; sy += w * rly; sz += w * rlz;
  }
  auto bred = [&](float v) -> float {
    red[t] = v; __syncthreads();
    for (int s = 128; s > 0; s >>= 1) { if (t < s) red[t] += red[t + s]; __syncthreads(); }
    float r = red[0]; __syncthreads();
    return r;
  };
  sum = bred(sum); sx = bred(sx); sy = bred(sy); sz = bred(sz);
  if (t == 0) {
    float inv = 1.0f / fmaxf(sum, 1e-6f);
    msgs[n * 3 + 0] = sx * inv;
    msgs[n * 3 + 1] = sy * inv;
    msgs[n * 3 + 2] = sz * inv;
  }
}

// ---------------------------------------------------------------------------
// 8) final projection [256->3], +0.25*msgs, rotate back through frame
__global__ void output_kernel(const float* __restrict__ hidden, const float* __restrict__ W_out2,
                              const float* __restrict__ b_out2, const float* __restrict__ msgs,
                              const float* __restrict__ frames, float* __restrict__ out) {
  int lane = threadIdx.x & 31;
  int n = blockIdx.x * (blockDim.x >> 5) + (threadIdx.x >> 5);
  if (n >= NN) return;
  float a0 = 0.f, a1 = 0.f, a2 = 0.f;
  for (int c = lane; c < HH; c += 32) {
    float v = hidden[n * HH + c];
    a0 += v * W_out2[c * 3 + 0];
    a1 += v * W_out2[c * 3 + 1];
    a2 += v * W_out2[c * 3 + 2];
  }
#pragma unroll
  for (int off = 16; off > 0; off >>= 1) {
    a0 += __shfl_xor(a0, off, 32);
    a1 += __shfl_xor(a1, off, 32);
    a2 += __shfl_xor(a2, off, 32);
  }
  if (lane == 0) {
    float v0 = a0 + b_out2[0] + 0.25f * msgs[n * 3 + 0];
    float v1 = a1 + b_out2[1] + 0.25f * msgs[n * 3 + 1];
    float v2 = a2 + b_out2[2] + 0.25f * msgs[n * 3 + 2];
    const float* Fr = frames + n * 9;
    out[n * 3 + 0] = Fr[0] * v0 + Fr[1] * v1 + Fr[2] * v2;   // einsum nij,nj->ni
    out[n * 3 + 1] = Fr[3] * v0 + Fr[4] * v1 + Fr[5] * v2;
    out[n * 3 + 2] = Fr[6] * v0 + Fr[7] * v1 + Fr[8] * v2;
  }
}

// ---------------------------------------------------------------------------
extern "C" void kernel_launch(void* const* d_in, const int* in_sizes, int n_in,
                              void* d_out, int out_size, void* d_ws, size_t ws_size,
                              hipStream_t stream) {
  const float* h      = (const float*)d_in[0];
  const float* x_t    = (const float*)d_in[1];
  const float* x_cond = (const float*)d_in[2];
  const float* tau    = (const float*)d_in[3];
  const float* W_np1  = (const float*)d_in[4];
  const float* b_np1  = (const float*)d_in[5];
  const float* W_np2  = (const float*)d_in[6];
  const float* b_np2  = (const float*)d_in[7];
  const float* w_src  = (const float*)d_in[8];
  const float* b_src  = (const float*)d_in[9];
  const float* w_dst  = (const float*)d_in[10];
  const float* b_dst  = (const float*)d_in[11];
  const float* w_ns   = (const float*)d_in[12];
  const float* b_ns   = (const float*)d_in[13];
  const float* W_eg1  = (const float*)d_in[14];
  const float* b_eg1  = (const float*)d_in[15];
  const float* w_eg2  = (const float*)d_in[16];
  const float* b_eg2  = (const float*)d_in[17];
  const float* W_out1 = (const float*)d_in[18];
  const float* b_out1 = (const float*)d_in[19];
  const float* W_out2 = (const float*)d_in[20];
  const float* b_out2 = (const float*)d_in[21];
  float* out = (float*)d_out;
  float* ws = (float*)d_ws;

  // workspace layout (float offsets)
  float* frames  = ws + 0;          //  9216
  float* dl      = ws + 9216;       //  3072
  float* nf      = ws + 12288;      //  270336 : node_feat [1024,264]
  float* Wt1     = ws + 282624;     //  67584  : W_np1^T padded  [256,264]
  float* Wt2     = ws + 350208;     //  65536  : W_np2^T         [256,256]
  float* Wt3     = ws + 415744;     //  67584  : W_out1^T padded [256,264]
  float* nh1     = ws + 483328;     //  262144
  float* nh      = ws + 745472;     //  262144
  float* srcv    = ws + 1007616;    //  1024
  float* dstv    = ws + 1008640;    //  1024
  float* nsv     = ws + 1009664;    //  1024
  float* logits  = ws + 1010688;    //  1048576
  float* msgs    = ws + 2059264;    //  3072
  float* uf      = ws + 2062336;    //  270336
  float* hidden  = ws + 2332672;    //  262144  (end: 2594816 floats ~ 10.4 MB)

  frames_kernel<<<4, 256, 0, stream>>>(x_cond, x_t, frames, dl);
  transpose_pad_kernel<<<(HH * KP + 255) / 256, 256, 0, stream>>>(W_np1, Wt1, HH + 6, KP, HH);
  transpose_pad_kernel<<<(HH * HH + 255) / 256, 256, 0, stream>>>(W_np2, Wt2, HH, HH, HH);
  transpose_pad_kernel<<<(HH * KP + 255) / 256, 256, 0, stream>>>(W_out1, Wt3, HH + 6, KP, HH);
  nodefeat_kernel<<<(NN * KP + 255) / 256, 256, 0, stream>>>(h, dl, tau, nf);

  // nh1 = silu(nf @ W_np1 + b_np1) : M=1024,N=256,K=264 -> 256 waves -> 32 blocks of 8 waves
  wmma_gemm_silu<<<32, 256, 0, stream>>>(nf, Wt1, b_np1, nh1, NN, HH, KP);
  // nh  = silu(nh1 @ W_np2 + b_np2): K=256
  wmma_gemm_silu<<<32, 256, 0, stream>>>(nh1, Wt2, b_np2, nh, NN, HH, HH);

  heads_kernel<<<128, 256, 0, stream>>>(nh, w_src, b_src, w_dst, b_dst, w_ns, b_ns,
                                        srcv, dstv, nsv);

  // 1024 waves (one per node), 4 waves/block
  edge_logits_kernel<<<256, 128, 0, stream>>>(x_t, frames, W_eg1, b_eg1, w_eg2, b_eg2,
                                              srcv, dstv, logits);

  softmax_msgs_kernel<<<NN, 256, 0, stream>>>(logits, x_t, frames, nsv, msgs);

  uf_kernel<<<(NN * KP + 255) / 256, 256, 0, stream>>>(nh, dl, msgs, uf);
  wmma_gemm_silu<<<32, 256, 0, stream>>>(uf, Wt3, b_out1, hidden, NN, HH, KP);

  output_kernel<<<128, 256, 0, stream>>>(hidden, W_out2, b_out2, msgs, frames, out);
}